// Cell_14654428414368
// MI455X (gfx1250) — compile-verified
//
#include <hip/hip_runtime.h>
#include <math.h>

// Problem constants (from reference)
#define N_NODES 50000
#define N_HID   128
#define N_ADJ   4
#define N_EDGE  600000
#define TOTAL_EDGES (N_ADJ * N_EDGE)   // 2,400,000

typedef __attribute__((ext_vector_type(2))) float v2f;
typedef __attribute__((ext_vector_type(8))) float v8f;

// Guaranteed native CDNA5 fire-and-forget FP32 atomic:
//   GLOBAL_ATOMIC_ADD_F32, non-returning (STOREcnt), RMW at device
//   scope so it resolves at the shared 192MB L2 (accumulator is L2-resident).
__device__ __forceinline__ void global_fadd_f32(float* p, float v) {
    asm volatile("global_atomic_add_f32 %0, %1, off scope:SCOPE_DEV"
                 :
                 : "v"(p), "v"(v)
                 : "memory");
}

// ---------------------------------------------------------------------------
// Kernel 1: zero the float accumulator in workspace (N_NODES*N_HID floats)
// ---------------------------------------------------------------------------
__global__ void zero_acc(float4* __restrict__ ws, int n4) {
    int i = blockIdx.x * blockDim.x + threadIdx.x;
    if (i < n4) {
        float4 z; z.x = 0.f; z.y = 0.f; z.z = 0.f; z.w = 0.f;
        ws[i] = z;
    }
}

// ---------------------------------------------------------------------------
// Kernel 2: edge scatter.  One wave (32 lanes) per edge.
//   For k in 0..3: lane L handles channel k*32+L -> both the gather load and
//   the f32 atomic hit one contiguous 128B cache line per instruction.
//   Edge metadata is single-use -> non-temporal loads keep x/acc L2-resident.
// ---------------------------------------------------------------------------
__global__ __launch_bounds__(256) void scatter_edges(
    const float* __restrict__ x,
    const float* __restrict__ weight,
    const int*   __restrict__ adj_src,
    const int*   __restrict__ adj_dst,
    const float* __restrict__ adj_vals,
    float*       __restrict__ acc)
{
    const int lane = threadIdx.x & 31;
    const int e = blockIdx.x * (blockDim.x >> 5) + (threadIdx.x >> 5);
    if (e >= TOTAL_EDGES) return;

    const int   s  = __builtin_nontemporal_load(adj_src + e);
    const int   d  = __builtin_nontemporal_load(adj_dst + e);
    const float v  = __builtin_nontemporal_load(adj_vals + e);
    const int   a  = e / N_EDGE;               // adjacency index
    const float sc = weight[a] * v;

    const float* xr   = x   + (long long)s * N_HID;
    float*       orow = acc + (long long)d * N_HID;

    #pragma unroll
    for (int k = 0; k < 4; ++k) {
        float m = xr[k * 32 + lane] * sc;
        global_fadd_f32(orow + k * 32 + lane, m);
    }
}

// ---------------------------------------------------------------------------
// Kernel 3: per-row L2 normalize + exact GELU.
//   One wave handles 16 nodes.  Sum-of-squares via V_WMMA_F32_16X16X4_F32:
//   the f32 A(16x4) and B(4x16) VGPR layouts coincide, so D = A * A^T and
//   diag(D)[m] = sum_k row_m[k]^2, accumulated over K=128 in 32 WMMAs.
// ---------------------------------------------------------------------------
__global__ __launch_bounds__(160) void finalize_norm_gelu(
    const float* __restrict__ acc, float* __restrict__ out)
{
    const int lane = threadIdx.x & 31;
    const int wave = blockIdx.x * (blockDim.x >> 5) + (threadIdx.x >> 5);
    if (wave >= (N_NODES / 16)) return;        // wave-uniform: EXEC stays full

    const int node   = wave * 16 + (lane & 15);
    const int colOff = 2 * (lane >> 4);        // lanes 0-15 -> {K0,K1}; 16-31 -> {K2,K3}
    const float* rowp = acc + (long long)node * N_HID + colOff;

    v2f vbuf[32];
    v8f c = {};
    #pragma unroll
    for (int j = 0; j < 32; ++j) {
        v2f a;
        a.x = rowp[4 * j];
        a.y = rowp[4 * j + 1];
        vbuf[j] = a;
        // D(16x16,f32) += A(16x4,f32) x B(4x16,f32); same regs => Gram matrix
        c = __builtin_amdgcn_wmma_f32_16x16x4_f32(
                false, a, false, a, (short)0, c, false, false);
    }

    // Extract diag(D): node m<8 lives at lane m comp m; m>=8 at lane m+16 comp m-8.
    const int sel = lane & 7;
    float dv = c[0];
    dv = (sel == 1) ? c[1] : dv;
    dv = (sel == 2) ? c[2] : dv;
    dv = (sel == 3) ? c[3] : dv;
    dv = (sel == 4) ? c[4] : dv;
    dv = (sel == 5) ? c[5] : dv;
    dv = (sel == 6) ? c[6] : dv;
    dv = (sel == 7) ? c[7] : dv;
    const int nm      = lane & 15;
    const int srcLane = (nm < 8) ? nm : (nm + 16);
    const float sumsq = __shfl(dv, srcLane, 32);

    // F.normalize: x / max(||x||, 1e-12)
    const float scale = 1.0f / fmaxf(sqrtf(sumsq), 1e-12f);

    float* op = out + (long long)node * N_HID + colOff;
    const float inv_sqrt2 = 0.70710678118654752440f;
    #pragma unroll
    for (int j = 0; j < 32; ++j) {
        float vx = vbuf[j].x * scale;
        float vy = vbuf[j].y * scale;
        // exact (erf) GELU
        vx = 0.5f * vx * (1.0f + erff(vx * inv_sqrt2));
        vy = 0.5f * vy * (1.0f + erff(vy * inv_sqrt2));
        op[4 * j]     = vx;
        op[4 * j + 1] = vy;
    }
}

// ---------------------------------------------------------------------------
extern "C" void kernel_launch(void* const* d_in, const int* in_sizes, int n_in,
                              void* d_out, int out_size, void* d_ws, size_t ws_size,
                              hipStream_t stream) {
    const float* x        = (const float*)d_in[0];
    const float* weight   = (const float*)d_in[1];
    const int*   adj_src  = (const int*)d_in[2];
    const int*   adj_dst  = (const int*)d_in[3];
    const float* adj_vals = (const float*)d_in[4];
    float*       out      = (float*)d_out;
    float*       acc      = (float*)d_ws;   // N_NODES*N_HID floats = 25.6 MB

    // 1) zero accumulator
    const int n4 = (N_NODES * N_HID) / 4;    // 1,600,000 float4
    zero_acc<<<(n4 + 255) / 256, 256, 0, stream>>>((float4*)acc, n4);

    // 2) edge scatter: one wave per edge, 8 waves / 256-thread block
    const int nblocks = (TOTAL_EDGES + 7) / 8;   // 300,000
    scatter_edges<<<nblocks, 256, 0, stream>>>(x, weight, adj_src, adj_dst,
                                               adj_vals, acc);

    // 3) normalize + GELU: 3125 waves (16 nodes each), 5 waves / 160-thread block
    const int tiles = N_NODES / 16;              // 3125
    finalize_norm_gelu<<<(tiles + 4) / 5, 160, 0, stream>>>(acc, out);
}